// TimestepNorm_68315749810464
// MI455X (gfx1250) — compile-verified
//
#include <hip/hip_runtime.h>

// Problem constants (from reference)
#define BB   4
#define LL   4096
#define DD   2048
#define GG   32
#define GSZ  64          // group size
#define TILE 64          // timesteps per tile
#define NTILE (LL / TILE)
#define ROWF 66          // padded LDS row stride in floats (TDM pad: +2 dwords / 64 dwords)
#define NBUF 3           // LDS tile buffers (triple buffering)
#define PFD  5           // software prefetch distance (tiles) into L2
#define EPSV 1e-5f

typedef __attribute__((ext_vector_type(2))) float    v2f;
typedef __attribute__((ext_vector_type(8))) float    v8f;
typedef __attribute__((ext_vector_type(4))) unsigned u32x4;
typedef __attribute__((ext_vector_type(8))) int      i32x8;
typedef __attribute__((ext_vector_type(4))) int      i32x4;

// Issue one TDM load: 2D tile (TILE rows x GSZ f32), row stride DD elements,
// into LDS at byte offset lds_off with 2-dword padding after each 64-dword row.
__device__ __forceinline__ void tdm_load_tile(unsigned lds_off, const float* gp)
{
    unsigned long long ga = (unsigned long long)(uintptr_t)gp;
    u32x4 g0;
    g0[0] = 1u;                                        // count=1, user descriptor
    g0[1] = lds_off;                                   // lds_addr (bytes)
    g0[2] = (unsigned)(ga & 0xFFFFFFFFu);              // global_addr[31:0]
    g0[3] = (unsigned)((ga >> 32) & 0x01FFFFFFu)       // global_addr[56:32]
          | (2u << 30);                                // type = 2 ("image")
    i32x8 g1;
    g1[0] = (2 << 16)      // data_size = 4 bytes
          | (1 << 20)      // pad_enable
          | (5 << 22)      // pad_interval: 64 dwords
          | (1 << 25);     // pad_amount: 2 dwords
    g1[1] = (GSZ  & 0xFFFF) << 16;   // tensor_dim0 (low 16) at bits 63:48
    g1[2] = (TILE & 0xFFFF) << 16;   // tensor_dim0 hi=0; tensor_dim1 low16 at 95:80
    g1[3] = (GSZ  & 0xFFFF) << 16;   // tensor_dim1 hi=0; tile_dim0 at 127:112
    g1[4] = (TILE & 0xFFFF);         // tile_dim1; tile_dim2 = 0
    g1[5] = DD;                      // tensor_dim0_stride low 32 (elements)
    g1[6] = 0;                       // stride hi; tensor_dim1_stride low = 0 (2D)
    g1[7] = 0;
    i32x4 z4 = {0, 0, 0, 0};
    i32x8 z8 = {0, 0, 0, 0, 0, 0, 0, 0};
    // amdgpu-toolchain (clang-23) form: 6 args, extra int32x8 group zero-filled.
    __builtin_amdgcn_tensor_load_to_lds(g0, g1, z4, z4, z8, 0);
}

__global__ __launch_bounds__(256)
void TimestepNorm_68315749810464_kernel(const float* __restrict__ x,
                                        const int*   __restrict__ prev_count,
                                        const float* __restrict__ prev_mean,
                                        const float* __restrict__ prev_var,
                                        const unsigned char* __restrict__ mask,
                                        const float* __restrict__ weight,
                                        const float* __restrict__ bias,
                                        float* __restrict__ out)
{
    const int bg   = blockIdx.x;
    const int b    = bg / GG;
    const int g    = bg % GG;
    const int tid  = threadIdx.x;
    const int lane = tid & 31;
    const int wave = tid >> 5;

    extern __shared__ float smem[];
    float* buf     = smem;                       // NBUF * TILE * ROWF
    float* mArr    = smem + NBUF * TILE * ROWF;  // TILE
    float* meanArr = mArr + TILE;                // TILE
    float* rstdArr = meanArr + TILE;             // TILE

    float* yOut     = out;
    float* outCnt   = out + (size_t)BB * LL * DD;
    float* outMean  = outCnt + BB;
    float* outVar   = outMean + BB * GG;

    // Per-lane column pair (constant across tiles): fully coalesced 256B rows.
    const int colbase = g * GSZ + 2 * lane;
    const float w0 = weight[colbase], w1 = weight[colbase + 1];
    const float bi0 = bias[colbase],  bi1 = bias[colbase + 1];

    // Scan carries (used by wave 0; lane-uniform there).
    float Ccnt = (float)prev_count[b];
    const float mu0  = prev_mean[(size_t)b * GG + g];
    const float var0 = prev_var[(size_t)b * GG + g];
    float Csum  = Ccnt * mu0;
    float Csumq = Ccnt * mu0 * mu0;
    const float M2base = var0 * fmaxf(Ccnt, 1.0f);

    const float*         xg = x    + (size_t)b * LL * DD + (size_t)g * GSZ;
    const unsigned char* mb = mask + (size_t)b * LL;
    const unsigned ldsBase = (unsigned)(uintptr_t)buf;   // LDS byte offset (low 32 bits)

    // Prologue: wave 0 issues NBUF-1 TDM tile loads.
    if (wave == 0) {
        for (int p = 0; p < NBUF - 1; ++p)
            tdm_load_tile(ldsBase + (unsigned)((p % NBUF) * TILE * ROWF * 4),
                          xg + (size_t)(p * TILE) * DD);
    }

    for (int i = 0; i < NTILE; ++i) {
        // Deep L2 prefetch of a future tile (global_prefetch_b8).
        {
            const int pf = i + PFD;
            if (pf < NTILE && tid < 128) {
                const int prow = tid >> 1;
                const int poff = (tid & 1) * 32;   // two 128B lines per 256B row
                __builtin_prefetch(xg + (size_t)(pf * TILE + prow) * DD + poff, 0, 3);
            }
        }

        if (wave == 0) {
            const int nx = i + NBUF - 1;           // next tile to fetch
            if (nx < NTILE)
                tdm_load_tile(ldsBase + (unsigned)((nx % NBUF) * TILE * ROWF * 4),
                              xg + (size_t)(nx * TILE) * DD);
            int younger = NTILE - 1 - i;
            if (younger > NBUF - 1) younger = NBUF - 1;
            switch (younger) {                      // TDM completes in order per wave
                case 2:  __builtin_amdgcn_s_wait_tensorcnt(2); break;
                case 1:  __builtin_amdgcn_s_wait_tensorcnt(1); break;
                default: __builtin_amdgcn_s_wait_tensorcnt(0); break;
            }
        }
        __syncthreads();

        float* tb = buf + (i % NBUF) * TILE * ROWF;

        // ---- Per-timestep group sums via V_WMMA_F32_16X16X4_F32 (waves 0..3) ----
        // A = 16 timesteps x 4 columns (f32), B = ones(4x16) -> every D column is
        // the row sum; 16 chained WMMAs cover all 64 columns.
        if (wave < 4) {
            const int rbase = wave * 16;
            const int arow  = rbase + (lane & 15);
            const int ksub  = (lane >> 4) * 2;     // lanes 16..31 hold K=2,3
            v8f acc = {};
            v2f ones = {1.0f, 1.0f};
            #pragma unroll
            for (int c = 0; c < 16; ++c) {
                v2f a = *(const v2f*)(tb + arow * ROWF + 4 * c + ksub);
                acc = __builtin_amdgcn_wmma_f32_16x16x4_f32(
                          false, a, false, ones, (short)0, acc, false, false);
            }
            if ((lane & 15) == 0) {                // lanes 0 / 16 hold rows 0..7 / 8..15
                const int r0 = rbase + (lane >> 4) * 8;
                #pragma unroll
                for (int r = 0; r < 8; ++r)
                    mArr[r0 + r] = acc[r] * (1.0f / GSZ);
            }
        }
        __syncthreads();

        // ---- Parallel Welford prefix over the 64 tile timesteps (wave 0) ----
        if (wave == 0) {
            const int t0 = 2 * lane, t1 = t0 + 1;
            const int trow = i * TILE;
            float m0 = mArr[t0], m1 = mArr[t1];
            float v0 = mb[trow + t0] ? 1.0f : 0.0f;
            float v1 = mb[trow + t1] ? 1.0f : 0.0f;
            float s0 = v0 * m0, s1 = v1 * m1;
            float q0 = s0 * m0, q1 = s1 * m1;
            float cp = v0 + v1, sp = s0 + s1, qp = q0 + q1;
            float ci = cp, si = sp, qi = qp;       // inclusive scan of lane pairs
            #pragma unroll
            for (int d = 1; d < 32; d <<= 1) {
                float cu = __shfl_up(ci, d, 32);
                float su = __shfl_up(si, d, 32);
                float qu = __shfl_up(qi, d, 32);
                if (lane >= d) { ci += cu; si += su; qi += qu; }
            }
            const float ce = ci - cp, se = si - sp, qe = qi - qp;  // exclusive
            {   // timestep t0
                float c  = Ccnt + ce + v0;
                float S  = Csum + se + s0;
                float Q  = Csumq + qe + q0;
                float cs = fmaxf(c, 1.0f);
                float mn = S / cs;
                float vr = fmaxf((M2base + Q - c * mn * mn) / cs, 0.0f);
                meanArr[t0] = mn;
                rstdArr[t0] = rsqrtf(vr + EPSV);
            }
            {   // timestep t1
                float c  = Ccnt + ce + cp;
                float S  = Csum + se + sp;
                float Q  = Csumq + qe + qp;
                float cs = fmaxf(c, 1.0f);
                float mn = S / cs;
                float vr = fmaxf((M2base + Q - c * mn * mn) / cs, 0.0f);
                meanArr[t1] = mn;
                rstdArr[t1] = rsqrtf(vr + EPSV);
            }
            Ccnt  += __shfl(ci, 31, 32);           // carry to next tile
            Csum  += __shfl(si, 31, 32);
            Csumq += __shfl(qi, 31, 32);
        }
        __syncthreads();

        // ---- Normalize & store (all 8 waves, 8 rows each; 256B coalesced rows) ----
        {
            const int rbase = wave * 8;
            #pragma unroll
            for (int r = 0; r < 8; ++r) {
                const int row = rbase + r;
                const float mn = meanArr[row];
                const float rs = rstdArr[row];
                const float* src = tb + row * ROWF + 2 * lane;
                float x0 = src[0], x1 = src[1];
                v2f o;
                o[0] = (x0 - mn) * rs * w0 + bi0;
                o[1] = (x1 - mn) * rs * w1 + bi1;
                *(v2f*)(yOut + (size_t)b * LL * DD
                             + (size_t)(i * TILE + row) * DD + colbase) = o;
            }
        }
        __syncthreads();   // tile buffer free for TDM reuse
    }

    // Final carries -> count / mean / var outputs.
    if (wave == 0 && lane == 0) {
        float cs = fmaxf(Ccnt, 1.0f);
        float mn = Csum / cs;
        float vr = (M2base + Csumq - Ccnt * mn * mn) / cs;
        outMean[(size_t)b * GG + g] = mn;
        outVar[(size_t)b * GG + g]  = vr;
        if (g == 0) outCnt[b] = Ccnt;   // int count stored as float value
    }
}

extern "C" void kernel_launch(void* const* d_in, const int* in_sizes, int n_in,
                              void* d_out, int out_size, void* d_ws, size_t ws_size,
                              hipStream_t stream) {
    (void)in_sizes; (void)n_in; (void)out_size; (void)d_ws; (void)ws_size;
    const float*         x  = (const float*)d_in[0];
    const int*           pc = (const int*)d_in[1];
    const float*         pm = (const float*)d_in[2];
    const float*         pv = (const float*)d_in[3];
    const unsigned char* mk = (const unsigned char*)d_in[4];
    const float*         w  = (const float*)d_in[5];
    const float*         bi = (const float*)d_in[6];

    const size_t smemBytes = (size_t)(NBUF * TILE * ROWF + 3 * TILE) * sizeof(float);
    TimestepNorm_68315749810464_kernel<<<dim3(BB * GG), dim3(256), smemBytes, stream>>>(
        x, pc, pm, pv, mk, w, bi, (float*)d_out);
}